// Prototypical_68642167325149
// MI455X (gfx1250) — compile-verified
//
#include <hip/hip_runtime.h>
#include <hip/hip_bf16.h>
#include <math.h>

// Problem sizes (fixed by the reference).
#define N_ROWS 4096
#define D_DIM  128
#define K_COLS 512
#define EPS_F  1e-6f

typedef __attribute__((ext_vector_type(2))) float v2f;
typedef __attribute__((ext_vector_type(8))) float v8f;

// ---------------------------------------------------------------------------
// Kernel 1: row norms of a' = net_out + eps  and column norms of w.
// ---------------------------------------------------------------------------
__global__ void proto_norms(const float* __restrict__ net_out,
                            const float* __restrict__ w,
                            float* __restrict__ rowNorm,
                            float* __restrict__ colNorm) {
    int i = blockIdx.x * blockDim.x + threadIdx.x;
    if (i < N_ROWS) {
        const float* p = net_out + (size_t)i * D_DIM;
        float s = 0.f;
        #pragma unroll 8
        for (int d = 0; d < D_DIM; ++d) {
            float v = p[d] + EPS_F;
            s = fmaf(v, v, s);
        }
        rowNorm[i] = s;
    } else if (i < N_ROWS + K_COLS) {
        int k = i - N_ROWS;
        float s = 0.f;
        #pragma unroll 8
        for (int d = 0; d < D_DIM; ++d) {
            float v = w[(size_t)d * K_COLS + k];
            s = fmaf(v, v, s);
        }
        colNorm[k] = s;
    }
}

// ---------------------------------------------------------------------------
// Kernel 2: WMMA f32 GEMM + distance epilogue.
//
// Block = 256 threads (8 wave32), grid = (K/16, N/256).
// Per block: stage the 128x16 f32 tile of w (8 KB) into LDS ONCE via
// GLOBAL_LOAD_ASYNC_TO_LDS_B128 (ASYNCcnt path), then each wave computes two
// 16x16 output tiles (rows w*32..w*32+31) with two independent WMMA
// accumulator chains sharing each B fragment.
//
// Fragment layouts (ISA 7.12.2, f32 16x16x4):
//   A: lanes 0-15 -> M=lane, {K0,K1}; lanes 16-31 -> M=lane-16, {K2,K3}
//   B: VGPR0 = rows K0/K2 (lane halves), VGPR1 = rows K1/K3, N = lane & 15
//   C: VGPR r -> row r + 8*(lane>=16), N = lane & 15
// ---------------------------------------------------------------------------
__global__ __launch_bounds__(256)
void proto_wmma_scores(const float* __restrict__ net_out,
                       const float* __restrict__ w,
                       const float* __restrict__ rowNorm,
                       const float* __restrict__ colNorm,
                       float* __restrict__ scores) {
    __shared__ float bTile[D_DIM * 16];          // [d][n], 8 KB

    const int t       = threadIdx.x;
    const int colBase = blockIdx.x * 16;
    const int rowBlk  = blockIdx.y * 256;

    // ---- Stage B tile: 2048 floats = 512 x 16B chunks, 2 per thread.
    // chunk c: d = c>>2, n4 = (c&3)*4  ->  w[d*512 + colBase + n4 .. +3]
    #pragma unroll
    for (int it = 0; it < 2; ++it) {
        const int c  = t + it * 256;
        const int d  = c >> 2;
        const int n4 = (c & 3) * 4;
        const unsigned ldsAddr = (unsigned)(size_t)(&bTile[c * 4]); // LDS byte addr
        const float* g = w + (size_t)d * K_COLS + colBase + n4;
        asm volatile("global_load_async_to_lds_b128 %0, %1, off"
                     :
                     : "v"(ldsAddr), "v"(g)
                     : "memory");
    }
#if __has_builtin(__builtin_amdgcn_s_wait_asynccnt)
    __builtin_amdgcn_s_wait_asynccnt(0);
#else
    asm volatile("s_wait_asynccnt 0x0" ::: "memory");
#endif
    __syncthreads();

    // ---- Per-wave compute: two 16x16 row tiles sharing B fragments.
    const int lane = t & 31;
    const int wave = t >> 5;
    const int m    = lane & 15;
    const int half = lane >> 4;
    const int koff = half * 2;

    const int row0 = rowBlk + wave * 32;         // first tile rows
    const int row1 = row0 + 16;                  // second tile rows

    const float* aRow0 = net_out + (size_t)(row0 + m) * D_DIM + koff;
    const float* aRow1 = net_out + (size_t)(row1 + m) * D_DIM + koff;

    v8f c0 = {0.f, 0.f, 0.f, 0.f, 0.f, 0.f, 0.f, 0.f};
    v8f c1 = {0.f, 0.f, 0.f, 0.f, 0.f, 0.f, 0.f, 0.f};

    #pragma unroll
    for (int kk = 0; kk < D_DIM / 4; ++kk) {
        const int k0 = 4 * kk;
        v2f a0, a1, b;
        a0.x = aRow0[k0]     + EPS_F;
        a0.y = aRow0[k0 + 1] + EPS_F;
        a1.x = aRow1[k0]     + EPS_F;
        a1.y = aRow1[k0 + 1] + EPS_F;
        b.x  = bTile[(k0 + koff)     * 16 + m];  // ds_load, conflict-free banks
        b.y  = bTile[(k0 + koff + 1) * 16 + m];
        // 8 args: (neg_a, A, neg_b, B, c_mod, C, reuse_a, reuse_b)
        c0 = __builtin_amdgcn_wmma_f32_16x16x4_f32(
                 false, a0, false, b, (short)0, c0, false, false);
        c1 = __builtin_amdgcn_wmma_f32_16x16x4_f32(
                 false, a1, false, b, (short)0, c1, false, false);
    }

    // ---- Epilogue: score = -sqrt(max(rn - 2*dot + cn, 0))
    const float cn = colNorm[colBase + m];
    #pragma unroll
    for (int r = 0; r < 8; ++r) {
        const int ra = row0 + r + 8 * half;
        const float rna = rowNorm[ra];
        float d2a = fmaxf(fmaf(-2.f, c0[r], rna + cn), 0.f);
        scores[(size_t)ra * K_COLS + colBase + m] = -sqrtf(d2a);

        const int rb = row1 + r + 8 * half;
        const float rnb = rowNorm[rb];
        float d2b = fmaxf(fmaf(-2.f, c1[r], rnb + cn), 0.f);
        scores[(size_t)rb * K_COLS + colBase + m] = -sqrtf(d2b);
    }
}

// ---------------------------------------------------------------------------
// Kernel 3: per-row CE term: terms[n] = logsumexp(scores[n,:]) - s[label[n]]
// ---------------------------------------------------------------------------
__global__ __launch_bounds__(128)
void proto_ce_rows(const float* __restrict__ scores,
                   const int* __restrict__ label,
                   float* __restrict__ terms) {
    const int n = blockIdx.x;
    const int t = threadIdx.x;
    const float* row = scores + (size_t)n * K_COLS;
    __shared__ float red[128];

    float mx = -3.402823466e38f;
    for (int k = t; k < K_COLS; k += 128) mx = fmaxf(mx, row[k]);
    red[t] = mx;
    __syncthreads();
    for (int s = 64; s > 0; s >>= 1) {
        if (t < s) red[t] = fmaxf(red[t], red[t + s]);
        __syncthreads();
    }
    mx = red[0];
    __syncthreads();

    float sum = 0.f;
    for (int k = t; k < K_COLS; k += 128) sum += __expf(row[k] - mx);
    red[t] = sum;
    __syncthreads();
    for (int s = 64; s > 0; s >>= 1) {
        if (t < s) red[t] += red[t + s];
        __syncthreads();
    }
    if (t == 0) {
        const float lse = mx + __logf(red[0]);
        terms[n] = lse - row[label[n]];
    }
}

// ---------------------------------------------------------------------------
// Kernel 4: loss = mean(terms) -> d_out[N*K]. Single block, fixed-order tree.
// ---------------------------------------------------------------------------
__global__ __launch_bounds__(256)
void proto_reduce_loss(const float* __restrict__ terms,
                       float* __restrict__ out_loss) {
    __shared__ float red[256];
    const int t = threadIdx.x;
    float s = 0.f;
    for (int i = t; i < N_ROWS; i += 256) s += terms[i];
    red[t] = s;
    __syncthreads();
    for (int k = 128; k > 0; k >>= 1) {
        if (t < k) red[t] += red[t + k];
        __syncthreads();
    }
    if (t == 0) out_loss[0] = red[0] * (1.0f / N_ROWS);
}

// ---------------------------------------------------------------------------
extern "C" void kernel_launch(void* const* d_in, const int* in_sizes, int n_in,
                              void* d_out, int out_size, void* d_ws, size_t ws_size,
                              hipStream_t stream) {
    const float* net_out = (const float*)d_in[0];
    const int*   label   = (const int*)d_in[1];
    const float* w       = (const float*)d_in[2];

    float* scores = (float*)d_out;                       // N*K fp32
    float* loss   = scores + (size_t)N_ROWS * K_COLS;    // 1 fp32

    float* ws      = (float*)d_ws;
    float* rowNorm = ws;                     // 4096
    float* colNorm = ws + N_ROWS;            // 512
    float* terms   = ws + N_ROWS + K_COLS;   // 4096

    // 1) norms
    {
        int total = N_ROWS + K_COLS;
        proto_norms<<<(total + 255) / 256, 256, 0, stream>>>(net_out, w, rowNorm, colNorm);
    }
    // 2) WMMA GEMM + distance epilogue: 8-wave blocks, async B staging in LDS
    {
        dim3 grid(K_COLS / 16, N_ROWS / 256);  // (32, 16)
        proto_wmma_scores<<<grid, 256, 0, stream>>>(net_out, w, rowNorm, colNorm, scores);
    }
    // 3) per-row CE terms
    proto_ce_rows<<<N_ROWS, 128, 0, stream>>>(scores, label, terms);
    // 4) mean -> loss
    proto_reduce_loss<<<1, 256, 0, stream>>>(terms, loss);
}